// CompactCenterLoss_11338713661721
// MI455X (gfx1250) — compile-verified
//
#include <hip/hip_runtime.h>
#include <hip/hip_bf16.h>
#include <math.h>

typedef __attribute__((ext_vector_type(2))) float v2f;
typedef __attribute__((ext_vector_type(8))) float v8f;

#define BN 4096
#define DN 512
#define CN 100
#define EPSF 1e-12f
#define IAML_MARGIN 5.0f

// ws layout in 32-bit words:
//  [0,     4096)  ni        : ||inputs[i]||^2
//  [4096,  8192)  pos_sum   : sum of same-class distances per row
//  [8192, 12288)  tot_sum   : sum of all distances per row
//  [12288,12416)  count     : class histogram (int)
//  [12416,12420)  scalars   : [0]=loss1_sum, [1]=prec_sum
#define WS_NI   0
#define WS_POS  4096
#define WS_TOT  8192
#define WS_CNT  12288
#define WS_SCAL 12416
#define WS_WORDS 12420

__global__ void k_init(float* __restrict__ ws) {
    int i = blockIdx.x * blockDim.x + threadIdx.x;
    if (i < WS_WORDS) ws[i] = 0.0f;
}

// Row norms + class histogram. One wave per row.
__global__ void k_norms(const float* __restrict__ inp, const int* __restrict__ tgt,
                        float* __restrict__ ws) {
    int wave = (blockIdx.x * blockDim.x + threadIdx.x) >> 5;   // 0..4095 (exact)
    int lane = threadIdx.x & 31;
    const float* a = inp + (size_t)wave * DN;
    float s = 0.0f;
    #pragma unroll
    for (int k = 0; k < 16; ++k) { float x = a[lane + 32 * k]; s += x * x; }
    #pragma unroll
    for (int m = 1; m < 32; m <<= 1) s += __shfl_xor(s, m, 32);
    if (lane == 0) {
        ws[WS_NI + wave] = s;
        atomicAdd((int*)(ws + WS_CNT) + tgt[wave], 1);
    }
}

// Triplet part: dist_ap/dist_an over the 100 classes. One wave per row.
__global__ void k_loss1(const float* __restrict__ inp, const int* __restrict__ tgt,
                        const float* __restrict__ cen, float* __restrict__ ws) {
    int wave = (blockIdx.x * blockDim.x + threadIdx.x) >> 5;   // row index
    int lane = threadIdx.x & 31;
    const float* a = inp + (size_t)wave * DN;
    float av[16];
    #pragma unroll
    for (int k = 0; k < 16; ++k) av[k] = a[lane + 32 * k];
    int ti = tgt[wave];
    const int* cnt = (const int*)(ws + WS_CNT);
    float ap = 0.0f, an = 3.4e38f;
    for (int c = 0; c < CN; ++c) {
        const float* cv = cen + (size_t)c * DN;
        float s = 0.0f;
        #pragma unroll
        for (int k = 0; k < 16; ++k) { float d = av[k] - cv[lane + 32 * k]; s += d * d; }
        #pragma unroll
        for (int m = 1; m < 32; m <<= 1) s += __shfl_xor(s, m, 32);
        float d = sqrtf(fmaxf(s, EPSF));
        if (c == ti)            ap = d;
        else if (cnt[c] > 0)    an = fminf(an, d);
    }
    if (lane == 0) {
        atomicAdd(ws + WS_SCAL + 0, fmaxf(ap - an, 0.0f));       // margin = 0
        atomicAdd(ws + WS_SCAL + 1, (an > ap) ? 1.0f : 0.0f);    // precision
    }
}

static __device__ __forceinline__ v8f wmma_f32(float2 a, float2 b, v8f c) {
    v2f av; av[0] = a.x; av[1] = a.y;
    v2f bv; bv[0] = b.x; bv[1] = b.y;
    return __builtin_amdgcn_wmma_f32_16x16x4_f32(
        /*neg_a=*/false, av, /*neg_b=*/false, bv,
        /*c_mod=*/(short)0, c, /*reuse_a=*/false, /*reuse_b=*/false);
}

// Fused Gram (WMMA f32 16x16x4) -> sqrt -> masked row-sum, 2x2 register-blocked:
// each wave computes a 32x32 output patch per pair, reusing every A/B fragment
// twice -> 0.5 VMEM issues per WMMA. 32 staged rows in exactly 64 KB LDS with an
// XOR bank swizzle (addr = r*512 + (k ^ ((r&15)<<2))) so b64 A-frag reads hit all
// 64 banks conflict-free.
__global__ __launch_bounds__(256) void k_gram(const float* __restrict__ inp,
                                              const int* __restrict__ tgt,
                                              float* __restrict__ ws) {
    __shared__ float sA[32 * 512];   // 64 KB
    const int tid = threadIdx.x;
    const int i0  = blockIdx.x * 32;

    for (int idx = tid; idx < 32 * DN; idx += 256) {
        int r = idx >> 9, col = idx & 511;
        sA[(r << 9) + (col ^ ((r & 15) << 2))] = inp[(size_t)(i0 + r) * DN + col];
    }
    __syncthreads();

    const int lane = tid & 31;
    const int wave = tid >> 5;
    const int n    = lane & 15;    // A row within tile == B column within tile
    const int h    = lane >> 4;    // half-wave select (K offset / M offset)
    const int xr   = n << 2;       // per-lane LDS swizzle key
    const int koff = 2 * h;

    const float* ni = ws + WS_NI;
    float niI0[8], niI1[8]; int tI0[8], tI1[8];
    #pragma unroll
    for (int v = 0; v < 8; ++v) {
        int r0 = i0 + v + 8 * h;
        niI0[v] = ni[r0];      tI0[v] = tgt[r0];
        niI1[v] = ni[r0 + 16]; tI1[v] = tgt[r0 + 16];
    }

    float acc_t0[8], acc_p0[8], acc_t1[8], acc_p1[8];
    #pragma unroll
    for (int v = 0; v < 8; ++v) { acc_t0[v]=0.f; acc_p0[v]=0.f; acc_t1[v]=0.f; acc_p1[v]=0.f; }

    const float* pA0 = &sA[(n << 9)];
    const float* pA1 = &sA[((n + 16) << 9)];

    const int jbase = blockIdx.y * 2048 + wave * 256;   // 256-col chunk per wave
    for (int jp = 0; jp < 8; ++jp) {
        int j0 = jbase + jp * 32;
        int J0 = j0 + n;
        int J1 = j0 + 16 + n;
        int   tJ0 = tgt[J0],  tJ1 = tgt[J1];
        float nJ0 = ni[J0],   nJ1 = ni[J1];
        const float* B0 = inp + (size_t)J0 * DN + koff;
        const float* B1 = inp + (size_t)J1 * DN + koff;

        // pull next pair's B rows toward near caches (global_prefetch_b8)
        int Jn = J0 + 32; Jn = (Jn > BN - 1) ? (BN - 1) : Jn;
        __builtin_prefetch(inp + (size_t)Jn * DN, 0, 3);
        __builtin_prefetch(inp + (size_t)(Jn + ((Jn + 16 > BN - 1) ? 0 : 16)) * DN, 0, 3);

        v8f c00 = {}, c01 = {}, c10 = {}, c11 = {};
        #pragma unroll 4
        for (int k = 0; k < DN; k += 4) {
            int ofs = (k ^ xr) + koff;          // swizzled dword offset within row
            float2 a0 = *(const float2*)(pA0 + ofs);   // ds_load_b64
            float2 a1 = *(const float2*)(pA1 + ofs);   // ds_load_b64
            float2 b0 = *(const float2*)(B0 + k);      // global_load_b64
            float2 b1 = *(const float2*)(B1 + k);      // global_load_b64
            c00 = wmma_f32(a0, b0, c00);
            c01 = wmma_f32(a0, b1, c01);
            c10 = wmma_f32(a1, b0, c10);
            c11 = wmma_f32(a1, b1, c11);
        }
        #pragma unroll
        for (int v = 0; v < 8; ++v) {
            float d;
            d = sqrtf(fmaxf(niI0[v] + nJ0 - 2.0f * c00[v], EPSF));
            acc_t0[v] += d; acc_p0[v] += (tI0[v] == tJ0) ? d : 0.0f;
            d = sqrtf(fmaxf(niI0[v] + nJ1 - 2.0f * c01[v], EPSF));
            acc_t0[v] += d; acc_p0[v] += (tI0[v] == tJ1) ? d : 0.0f;
            d = sqrtf(fmaxf(niI1[v] + nJ0 - 2.0f * c10[v], EPSF));
            acc_t1[v] += d; acc_p1[v] += (tI1[v] == tJ0) ? d : 0.0f;
            d = sqrtf(fmaxf(niI1[v] + nJ1 - 2.0f * c11[v], EPSF));
            acc_t1[v] += d; acc_p1[v] += (tI1[v] == tJ1) ? d : 0.0f;
        }
    }

    float* pos = ws + WS_POS;
    float* tot = ws + WS_TOT;
    #pragma unroll
    for (int v = 0; v < 8; ++v) {
        float t0 = acc_t0[v], p0 = acc_p0[v];
        float t1 = acc_t1[v], p1 = acc_p1[v];
        #pragma unroll
        for (int m = 1; m < 16; m <<= 1) {
            t0 += __shfl_xor(t0, m, 32); p0 += __shfl_xor(p0, m, 32);
            t1 += __shfl_xor(t1, m, 32); p1 += __shfl_xor(p1, m, 32);
        }
        if (n == 0) {
            int r0 = i0 + v + 8 * h;
            atomicAdd(&tot[r0], t0);      atomicAdd(&pos[r0], p0);
            atomicAdd(&tot[r0 + 16], t1); atomicAdd(&pos[r0 + 16], p1);
        }
    }
}

// Combine everything into the two output scalars.
__global__ void k_final(const int* __restrict__ tgt, const float* __restrict__ ws,
                        float* __restrict__ out) {
    __shared__ float red[256];
    int tid = threadIdx.x;
    const float* pos = ws + WS_POS;
    const float* tot = ws + WS_TOT;
    const int*   cnt = (const int*)(ws + WS_CNT);
    float s = 0.0f;
    for (int i = tid; i < BN; i += 256) {
        int   t  = tgt[i];
        int   cp = cnt[t];
        float pm = pos[i] / (float)cp;
        int   ng = BN - cp;
        float nm = (tot[i] - pos[i]) / (float)(ng > 0 ? ng : 1);
        s += pm + fmaxf(IAML_MARGIN - nm, 0.0f);
    }
    red[tid] = s;
    __syncthreads();
    for (int st = 128; st > 0; st >>= 1) {
        if (tid < st) red[tid] += red[tid + st];
        __syncthreads();
    }
    if (tid == 0) {
        float loss2 = red[0] / (float)BN;
        float loss1 = ws[WS_SCAL + 0] / (float)BN;
        float prec  = ws[WS_SCAL + 1] / (float)BN;
        out[0] = loss1 + 0.5f * loss2;
        out[1] = prec;
    }
}

extern "C" void kernel_launch(void* const* d_in, const int* in_sizes, int n_in,
                              void* d_out, int out_size, void* d_ws, size_t ws_size,
                              hipStream_t stream) {
    const float* inp = (const float*)d_in[0];
    const int*   tgt = (const int*)d_in[1];
    const float* cen = (const float*)d_in[2];
    float* out = (float*)d_out;
    float* ws  = (float*)d_ws;

    k_init <<<(WS_WORDS + 255) / 256, 256, 0, stream>>>(ws);
    k_norms<<<BN / 8,  256, 0, stream>>>(inp, tgt, ws);          // 8 waves/block
    k_loss1<<<BN / 8,  256, 0, stream>>>(inp, tgt, cen, ws);
    k_gram <<<dim3(BN / 32, 2), 256, 0, stream>>>(inp, tgt, ws); // 32-row tiles
    k_final<<<1,       256, 0, stream>>>(tgt, ws, out);
}